// PointNetSetAbstraction_14963666059793
// MI455X (gfx1250) — compile-verified
//
#include <hip/hip_runtime.h>
#include <hip/hip_bf16.h>

#define BB 8
#define NN 16384
#define DD 64
#define SS 1024
#define KK 64
#define PTOT (BB*SS*KK)      /* 524288 pixels */
#define RAD2 0.09f
#define BN_EPS 1e-5f

typedef float v2f __attribute__((ext_vector_type(2)));
typedef float v8f __attribute__((ext_vector_type(8)));
typedef int   async_b128_t __attribute__((vector_size(16)));   // matches builtin param

// gfx1250 async global->LDS builtins (probe via __has_builtin; sync fallback).
#if defined(__has_builtin)
#  if __has_builtin(__builtin_amdgcn_global_load_async_to_lds_b128) && \
      __has_builtin(__builtin_amdgcn_s_wait_asynccnt)
#    define HAVE_ASYNC_LDS 1
#  endif
#endif
#ifndef HAVE_ASYNC_LDS
#  define HAVE_ASYNC_LDS 0
#endif

// ---------------------------------------------------------------------------
// 1) Farthest point sampling: one block per batch, 1024 threads, 16 pts/thread.
// ---------------------------------------------------------------------------
__global__ void fps_kernel(const float* __restrict__ xyz, const int* __restrict__ idx_in,
                           int* __restrict__ fps_idx, float* __restrict__ new_xyz,
                           int* __restrict__ new_ptr, float* __restrict__ out_xyz)
{
  const int b   = blockIdx.x;
  const int tid = threadIdx.x;                 // 1024 threads
  const float* xb = xyz + (size_t)b * 3 * NN;

  __shared__ int   s_fidx[SS];
  __shared__ float s_best[32];
  __shared__ int   s_bidx[32];
  __shared__ float s_c[3];
  __shared__ int   s_far;

  float dist[16];
  #pragma unroll
  for (int i = 0; i < 16; i++) dist[i] = 1e10f;

  if (tid == 0) s_far = 0;
  __syncthreads();

  for (int it = 0; it < SS; it++) {
    const int far = s_far;
    if (tid == 0) {
      fps_idx[b*SS + it] = far;
      s_fidx[it] = far;
      s_c[0] = xb[far]; s_c[1] = xb[NN + far]; s_c[2] = xb[2*NN + far];
    }
    __syncthreads();
    const float cx = s_c[0], cy = s_c[1], cz = s_c[2];

    float bd = -1.0f; int bi = 0;
    #pragma unroll
    for (int i = 0; i < 16; i++) {
      const int n = tid + i*1024;
      const float dx = xb[n] - cx, dy = xb[NN+n] - cy, dz = xb[2*NN+n] - cz;
      float d = dx*dx + dy*dy + dz*dz;
      d = fminf(dist[i], d);
      dist[i] = d;
      if (d > bd) { bd = d; bi = n; }          // n ascending -> first max kept
    }
    #pragma unroll
    for (int off = 1; off < 32; off <<= 1) {
      const float od = __shfl_xor(bd, off, 32);
      const int   oi = __shfl_xor(bi, off, 32);
      if (od > bd || (od == bd && oi < bi)) { bd = od; bi = oi; }
    }
    if ((tid & 31) == 0) { s_best[tid >> 5] = bd; s_bidx[tid >> 5] = bi; }
    __syncthreads();
    if (tid < 32) {
      bd = s_best[tid]; bi = s_bidx[tid];
      #pragma unroll
      for (int off = 1; off < 32; off <<= 1) {
        const float od = __shfl_xor(bd, off, 32);
        const int   oi = __shfl_xor(bi, off, 32);
        if (od > bd || (od == bd && oi < bi)) { bd = od; bi = oi; }
      }
      if (tid == 0) s_far = bi;
    }
    __syncthreads();
  }

  const int s = tid;
  const int n = s_fidx[s];
  const float x = xb[n], y = xb[NN+n], z = xb[2*NN+n];
  float* nx = new_xyz + ((size_t)b*SS + s)*3;
  nx[0] = x; nx[1] = y; nx[2] = z;
  out_xyz[(size_t)b*3*SS + 0*SS + s] = x;
  out_xyz[(size_t)b*3*SS + 1*SS + s] = y;
  out_xyz[(size_t)b*3*SS + 2*SS + s] = z;
  new_ptr[b*SS + s] = idx_in[b*NN + n];
}

// ---------------------------------------------------------------------------
// 2) Ball query: one wave per centroid; first NSAMPLE in-radius by index.
// ---------------------------------------------------------------------------
__global__ void ball_query_kernel(const float* __restrict__ xyz, const int* __restrict__ idx_in,
                                  const float* __restrict__ new_xyz, const int* __restrict__ new_ptr,
                                  int* __restrict__ gid)
{
  const int gw   = (int)((blockIdx.x * blockDim.x + threadIdx.x) >> 5);  // 0..8191
  const int lane = threadIdx.x & 31;
  const int b    = gw >> 10;
  const float* xb = xyz + (size_t)b * 3 * NN;
  const float cx = new_xyz[(size_t)gw*3 + 0];
  const float cy = new_xyz[(size_t)gw*3 + 1];
  const float cz = new_xyz[(size_t)gw*3 + 2];
  const float cn2 = cx*cx + cy*cy + cz*cz;
  const int np = new_ptr[gw];
  const int* ib = idx_in + (size_t)b * NN;
  int* g = gid + (size_t)gw * KK;

  int cnt = 0, first = 0;
  for (int n0 = 0; n0 < NN; n0 += 32) {
    const int n = n0 + lane;
    const float x = xb[n], y = xb[NN+n], z = xb[2*NN+n];
    const float d = cn2 + (x*x + y*y + z*z) - 2.0f*(cx*x + cy*y + cz*z);
    const bool ok = (ib[n] == np) && (d <= RAD2);
    const unsigned m = (unsigned)__ballot(ok);
    if (m) {
      if (cnt == 0) first = n0 + __builtin_ctz(m);
      if (ok) {
        const int pos = cnt + __builtin_popcount(m & ((1u << lane) - 1u));
        if (pos < KK) g[pos] = n;
      }
      cnt += __builtin_popcount(m);
      if (cnt >= KK) break;
    }
  }
  if (cnt < KK) {
    for (int p = cnt + lane; p < KK; p += 32) g[p] = first;
  }
}

// ---------------------------------------------------------------------------
// 3) Build layer-0 activation X0 [68, P]
// ---------------------------------------------------------------------------
__global__ void build_x0_kernel(const float* __restrict__ xyz, const float* __restrict__ points,
                                const float* __restrict__ new_xyz, const int* __restrict__ gid,
                                float* __restrict__ X0)
{
  const size_t p  = (size_t)blockIdx.x * blockDim.x + threadIdx.x;
  const size_t bs = p >> 6;
  const int b = (int)(bs >> 10);
  const int n = gid[p];
  const float* xb = xyz + (size_t)b * 3 * NN;
  const float* nx = new_xyz + bs * 3;
  X0[0*(size_t)PTOT + p] = xb[n]        - nx[0];
  X0[1*(size_t)PTOT + p] = xb[NN + n]   - nx[1];
  X0[2*(size_t)PTOT + p] = xb[2*NN + n] - nx[2];
  const float* pb = points + (size_t)b * DD * NN;
  #pragma unroll 4
  for (int c = 0; c < DD; c++)
    X0[(size_t)(3 + c) * PTOT + p] = pb[(size_t)c * NN + n];
  X0[(size_t)67 * PTOT + p] = 0.0f;                                  // K-pad row
}

__global__ void pad_w0_kernel(const float* __restrict__ w0, float* __restrict__ w0p)
{
  const int t = blockIdx.x * blockDim.x + threadIdx.x;
  if (t >= 128*68) return;
  const int oc = t / 68, c = t % 68;
  w0p[t] = (c < 67) ? w0[oc*67 + c] : 0.0f;
}

__global__ void zero_kernel(float* __restrict__ p, int n)
{
  const int t = blockIdx.x * blockDim.x + threadIdx.x;
  if (t < n) p[t] = 0.0f;
}

// ---------------------------------------------------------------------------
// 4) Async tile stage helper (global -> LDS, b128 per lane, ASYNCcnt tracked)
// ---------------------------------------------------------------------------
#if HAVE_ASYNC_LDS
template<int CIN, int NT>
__device__ __forceinline__ void async_stage(const float* __restrict__ X, size_t P,
                                            size_t px0, float* buf, int tid)
{
  constexpr int TRIPS = (CIN*4)/NT;            // b128 transfers per thread
  #pragma unroll
  for (int jj = 0; jj < TRIPS; jj++) {
    const int j = tid + jj*NT;
    const int row = j >> 2, c4 = j & 3;
    const float* gp = X + (size_t)row * P + px0 + c4*4;
    __builtin_amdgcn_global_load_async_to_lds_b128(
        (__attribute__((address_space(1))) async_b128_t*)(void*)gp,
        (__attribute__((address_space(3))) async_b128_t*)(void*)(buf + row*16 + c4*4),
        0, 0);
  }
}
#endif

// ---------------------------------------------------------------------------
// 5) Fused GEMM (fp32 WMMA 16x16x4) + input BN/ReLU + per-channel stats
//    (+ optional fused raw max/min pooling over K for the last layer).
// ---------------------------------------------------------------------------
template<int CIN, int NW, bool ASYNC_T, bool POOL>
__global__ void __launch_bounds__(NW*32)
gemm_bn_wmma_kernel(const float* __restrict__ X, const float* __restrict__ W,
                    float* __restrict__ Y,
                    const float* __restrict__ scale, const float* __restrict__ shift,
                    float* __restrict__ gsum, float* __restrict__ gsq,
                    float* __restrict__ pmax, float* __restrict__ pmin)
{
  constexpr int NT  = NW * 32;
  constexpr int OC  = NW * 16;
  constexpr int K4  = CIN / 4;
  constexpr int PXI = 32;                        // pixel tiles per block
  constexpr bool USE_ASYNC = ASYNC_T && (HAVE_ASYNC_LDS != 0) && (((CIN*4) % NT) == 0);
  constexpr int NBUF = USE_ASYNC ? 2 : 1;
  const int tid  = threadIdx.x;
  const int lane = tid & 31;
  const int wave = tid >> 5;
  const int l15  = lane & 15;
  const int hi   = lane >> 4;
  const int oc0  = wave * 16;
  const size_t P = (size_t)PTOT;

  __shared__ float sx[NBUF][CIN * 16];
  __shared__ float scoef[CIN], hcoef[CIN];
  __shared__ float ssum[OC], ssq[OC];

  const bool has_tf = (scale != nullptr);
  if (has_tf) {
    for (int i = tid; i < CIN; i += NT) { scoef[i] = scale[i]; hcoef[i] = shift[i]; }
  }

  // Preload per-wave A fragments: A[M=l15][K=hi*2+{0,1}] = W[oc0+l15][k]
  v2f a[K4];
  {
    const float* wr = W + (size_t)(oc0 + l15) * CIN + hi*2;
    #pragma unroll
    for (int k4 = 0; k4 < K4; k4++) { a[k4].x = wr[k4*4 + 0]; a[k4].y = wr[k4*4 + 1]; }
  }

  float rs1[8], rs2[8], mx[8], mn[8];
  #pragma unroll
  for (int r = 0; r < 8; r++) { rs1[r] = 0.0f; rs2[r] = 0.0f; mx[r] = -3.402823466e38f; mn[r] = 3.402823466e38f; }

  const size_t pxbase = (size_t)blockIdx.x * PXI * 16;

#if HAVE_ASYNC_LDS
  if constexpr (USE_ASYNC) {
    async_stage<CIN, NT>(X, P, pxbase, &sx[0][0], tid);   // prefetch tile 0
  }
#endif

  for (int t = 0; t < PXI; t++) {
    const int cur = USE_ASYNC ? (t & 1) : 0;
    const size_t px0 = pxbase + (size_t)t * 16;

#if HAVE_ASYNC_LDS
    if constexpr (USE_ASYNC) {
      constexpr int TRIPS = (CIN*4)/NT;
      if (t + 1 < PXI) {                         // overlap next-tile DMA with compute
        async_stage<CIN, NT>(X, P, pxbase + (size_t)(t+1)*16, &sx[(t+1) & 1][0], tid);
        __builtin_amdgcn_s_wait_asynccnt(TRIPS); // tile t landed (in-order), t+1 in flight
      } else {
        __builtin_amdgcn_s_wait_asynccnt(0);
      }
      __syncthreads();                           // publish all waves' tile-t data
      if (has_tf) {                              // in-place BN+ReLU on the tile
        for (int i = tid; i < CIN*16; i += NT) {
          const int row = i >> 4;
          sx[cur][i] = fmaxf(fmaf(sx[cur][i], scoef[row], hcoef[row]), 0.0f);
        }
        __syncthreads();
      }
    } else
#endif
    {
      __syncthreads();
      for (int i = tid; i < CIN*16; i += NT) {
        const int row = i >> 4, col = i & 15;
        float v = X[(size_t)row * P + px0 + col];
        if (has_tf) v = fmaxf(fmaf(v, scoef[row], hcoef[row]), 0.0f);
        sx[0][i] = v;
      }
      __syncthreads();
    }

    const float* bx = &sx[cur][0];
    v8f acc = {0.f,0.f,0.f,0.f,0.f,0.f,0.f,0.f};
    #pragma unroll
    for (int k4 = 0; k4 < K4; k4++) {
      v2f bb;                                    // B[K=hi*2+{0,1}][N=l15]
      bb.x = bx[(k4*4 + hi*2 + 0)*16 + l15];
      bb.y = bx[(k4*4 + hi*2 + 1)*16 + l15];
      acc = __builtin_amdgcn_wmma_f32_16x16x4_f32(false, a[k4], false, bb,
                                                  (short)0, acc, false, false);
    }

    if constexpr (!POOL) {
      float* yo = Y + (size_t)(oc0 + hi*8) * P + px0 + l15;
      #pragma unroll
      for (int r = 0; r < 8; r++) yo[(size_t)r * P] = acc[r];
    }
    #pragma unroll
    for (int r = 0; r < 8; r++) {
      const float v = acc[r];
      rs1[r] += v;
      rs2[r] += v * v;
    }
    if constexpr (POOL) {
      #pragma unroll
      for (int r = 0; r < 8; r++) { mx[r] = fmaxf(mx[r], acc[r]); mn[r] = fminf(mn[r], acc[r]); }
      if ((t & 3) == 3) {                        // one (b,s) group = 4 tiles done
        const size_t bs = (size_t)blockIdx.x * (PXI/4) + (size_t)(t >> 2);
        #pragma unroll
        for (int r = 0; r < 8; r++) {
          float vx = mx[r], vn = mn[r];
          #pragma unroll
          for (int off = 1; off < 16; off <<= 1) {
            vx = fmaxf(vx, __shfl_xor(vx, off, 32));
            vn = fminf(vn, __shfl_xor(vn, off, 32));
          }
          if (l15 == 0) {
            pmax[(size_t)(oc0 + hi*8 + r) * (BB*SS) + bs] = vx;
            pmin[(size_t)(oc0 + hi*8 + r) * (BB*SS) + bs] = vn;
          }
          mx[r] = -3.402823466e38f; mn[r] = 3.402823466e38f;
        }
      }
    }
#if HAVE_ASYNC_LDS
    if constexpr (USE_ASYNC) __syncthreads();    // safe buffer reuse next iter
#endif
  }

  // per-channel sums across the 16 lanes of each half-wave
  #pragma unroll
  for (int r = 0; r < 8; r++) {
    float s1 = rs1[r], s2 = rs2[r];
    #pragma unroll
    for (int off = 1; off < 16; off <<= 1) {
      s1 += __shfl_xor(s1, off, 32);
      s2 += __shfl_xor(s2, off, 32);
    }
    if (l15 == 0) {
      ssum[oc0 + hi*8 + r] = s1;
      ssq [oc0 + hi*8 + r] = s2;
    }
  }
  __syncthreads();
  if (tid < OC) {
    atomicAdd(&gsum[tid], ssum[tid]);
    atomicAdd(&gsq [tid], ssq [tid]);
  }
}

// ---------------------------------------------------------------------------
// 6) BN finalize: per-channel scale/shift (conv bias cancels under BN).
// ---------------------------------------------------------------------------
__global__ void bn_finalize_kernel(const float* __restrict__ sum, const float* __restrict__ sq,
                                   const float* __restrict__ g, const float* __restrict__ be,
                                   float* __restrict__ sc, float* __restrict__ sh, int oc)
{
  const int c = blockIdx.x * blockDim.x + threadIdx.x;
  if (c >= oc) return;
  const float inv = 1.0f / (float)PTOT;
  const float mu  = sum[c] * inv;
  float var = sq[c] * inv - mu*mu;
  var = fmaxf(var, 0.0f);
  const float r = rsqrtf(var + BN_EPS);
  const float s = g[c] * r;
  sc[c] = s;
  sh[c] = be[c] - mu * s;
}

// ---------------------------------------------------------------------------
// 7) Pool finish: out = relu(max(sc*mx+sh, sc*mn+sh))  (exact for any sc sign)
// ---------------------------------------------------------------------------
__global__ void pool_finish_kernel(const float* __restrict__ pmax, const float* __restrict__ pmin,
                                   const float* __restrict__ sc, const float* __restrict__ sh,
                                   float* __restrict__ out)
{
  const size_t t = (size_t)blockIdx.x * blockDim.x + threadIdx.x;   // b*256*S + c*S + s
  const int s = (int)(t & (SS - 1));
  const int c = (int)((t >> 10) & 255);
  const int b = (int)(t >> 18);
  const size_t bs = (size_t)b * SS + s;
  const float S_ = sc[c], H = sh[c];
  const float vx = fmaf(pmax[(size_t)c * (BB*SS) + bs], S_, H);
  const float vn = fmaf(pmin[(size_t)c * (BB*SS) + bs], S_, H);
  out[t] = fmaxf(0.0f, fmaxf(vx, vn));
}

// ---------------------------------------------------------------------------
extern "C" void kernel_launch(void* const* d_in, const int* in_sizes, int n_in,
                              void* d_out, int out_size, void* d_ws, size_t ws_size,
                              hipStream_t stream)
{
  (void)in_sizes; (void)n_in; (void)out_size; (void)ws_size;
  const float* xyz    = (const float*)d_in[0];
  const float* points = (const float*)d_in[1];
  const int*   idx    = (const int*)  d_in[2];
  const float* w0  = (const float*)d_in[3];
  const float* g0  = (const float*)d_in[5];
  const float* be0 = (const float*)d_in[6];
  const float* w1  = (const float*)d_in[7];
  const float* g1  = (const float*)d_in[9];
  const float* be1 = (const float*)d_in[10];
  const float* w2  = (const float*)d_in[11];
  const float* g2  = (const float*)d_in[13];
  const float* be2 = (const float*)d_in[14];
  float* out = (float*)d_out;

  char* base = (char*)d_ws;
  size_t off = 0;
  auto walloc = [&](size_t bytes) -> char* {
    char* p = base + off;
    off = (off + bytes + 255) & ~(size_t)255;
    return p;
  };
  int*   fps_idx = (int*)  walloc((size_t)BB*SS*sizeof(int));
  int*   new_ptr = (int*)  walloc((size_t)BB*SS*sizeof(int));
  float* new_xyz = (float*)walloc((size_t)BB*SS*3*sizeof(float));
  int*   gid     = (int*)  walloc((size_t)PTOT*sizeof(int));
  float* w0p     = (float*)walloc(128*68*sizeof(float));
  float* stats   = (float*)walloc(1024*sizeof(float));
  float* sum0 = stats;       float* sq0 = stats + 128;
  float* sum1 = stats + 256; float* sq1 = stats + 384;
  float* sum2 = stats + 512; float* sq2 = stats + 768;
  float* scsh = (float*)walloc(6*256*sizeof(float));
  float* sc0 = scsh;         float* sh0 = scsh + 256;
  float* sc1 = scsh + 512;   float* sh1 = scsh + 768;
  float* sc2 = scsh + 1024;  float* sh2 = scsh + 1280;
  float* pmax = (float*)walloc((size_t)256*BB*SS*sizeof(float));
  float* pmin = (float*)walloc((size_t)256*BB*SS*sizeof(float));
  float* Y0 = (float*)walloc((size_t)128*PTOT*sizeof(float));
  float* XY = (float*)walloc((size_t)128*PTOT*sizeof(float)); // X0 (68 rows), later Y1 (128)
  float* X0 = XY;
  float* Y1 = XY;                                             // X0 dead once layer-0 done

  fps_kernel<<<BB, 1024, 0, stream>>>(xyz, idx, fps_idx, new_xyz, new_ptr, out);
  ball_query_kernel<<<(BB*SS)/8, 256, 0, stream>>>(xyz, idx, new_xyz, new_ptr, gid);
  build_x0_kernel<<<PTOT/256, 256, 0, stream>>>(xyz, points, new_xyz, gid, X0);
  pad_w0_kernel<<<(128*68 + 255)/256, 256, 0, stream>>>(w0, w0p);
  zero_kernel<<<4, 256, 0, stream>>>(stats, 1024);

  gemm_bn_wmma_kernel< 68, 8, false, false><<<PTOT/(16*32), 256, 0, stream>>>(
      X0, w0p, Y0, nullptr, nullptr, sum0, sq0, nullptr, nullptr);
  bn_finalize_kernel<<<1, 128, 0, stream>>>(sum0, sq0, g0, be0, sc0, sh0, 128);

  gemm_bn_wmma_kernel<128, 8, true, false><<<PTOT/(16*32), 256, 0, stream>>>(
      Y0, w1, Y1, sc0, sh0, sum1, sq1, nullptr, nullptr);
  bn_finalize_kernel<<<1, 128, 0, stream>>>(sum1, sq1, g1, be1, sc1, sh1, 128);

  gemm_bn_wmma_kernel<128, 16, true, true><<<PTOT/(16*32), 512, 0, stream>>>(
      Y1, w2, nullptr, sc1, sh1, sum2, sq2, pmax, pmin);
  bn_finalize_kernel<<<1, 256, 0, stream>>>(sum2, sq2, g2, be2, sc2, sh2, 256);

  pool_finish_kernel<<<(BB*256*SS)/256, 256, 0, stream>>>(pmax, pmin, sc2, sh2, out + (size_t)BB*3*SS);
}